// MultiHeadAttention_61194694034288
// MI455X (gfx1250) — compile-verified
//
#include <hip/hip_runtime.h>
#include <hip/hip_bf16.h>

// ---------------- problem constants ----------------
#define EMB   1024
#define NH    16
#define DR    64      // EMB / NH
#define BB    8
#define SS    1024
#define MM    (BB * SS)          // 8192 rows for the big GEMMs
#define PLANE ((size_t)SS * SS)  // one (b,h) attention plane

#define QT    16                 // query rows per fused workgroup
#define KT    32                 // key cols per fused iteration

typedef __attribute__((ext_vector_type(2))) float v2f;
typedef __attribute__((ext_vector_type(8))) float v8f;

// D = A(16x4) * B(4x16) + C  via V_WMMA_F32_16X16X4_F32
__device__ __forceinline__ v8f wmma4(v2f a, v2f b, v8f c) {
    return __builtin_amdgcn_wmma_f32_16x16x4_f32(
        /*neg_a=*/false, a, /*neg_b=*/false, b,
        /*c_mod=*/(short)0, c, /*reuse_a=*/false, /*reuse_b=*/false);
}

// Fragment where the K dimension is contiguous in memory (row-major rows).
// Works for the A matrix, and for B in NT gemms (B[k][n] = W[n*ld + k]).
// Lane l holds elements K = kk + 2*(l>>4) + {0,1} of row (row0 + (l&15)).
__device__ __forceinline__ v2f frag_kcontig(const float* __restrict__ p, int ld,
                                            int row0, int kk, int lane) {
    const float* q = p + (size_t)(row0 + (lane & 15)) * ld + (kk + ((lane >> 4) << 1));
    v2f r; r.x = q[0]; r.y = q[1];
    return r;
}

// B fragment for NN gemms: B[k][n] = p[k*ld + n], K strided.
__device__ __forceinline__ v2f frag_b_nn(const float* __restrict__ p, int ld,
                                         int col0, int kk, int lane) {
    int n  = col0 + (lane & 15);
    int k0 = kk + ((lane >> 4) << 1);
    v2f r;
    r.x = p[(size_t)k0 * ld + n];
    r.y = p[(size_t)(k0 + 1) * ld + n];
    return r;
}

__device__ __forceinline__ v8f zero8() {
    v8f z = {0.f, 0.f, 0.f, 0.f, 0.f, 0.f, 0.f, 0.f};
    return z;
}

// ---------------------------------------------------------------------------
// Kernel 1: C[M,N] = A[M,K] * W[N,K]^T + bias   (torch Linear)
// block = 128 threads = 4 waves (2x2 of 64x64 wave tiles) -> 128x128 block tile
// ---------------------------------------------------------------------------
__global__ __launch_bounds__(128)
void gemm_nt_bias_k(const float* __restrict__ A, const float* __restrict__ W,
                    const float* __restrict__ bias, float* __restrict__ C,
                    int N, int K) {
    const int lane = threadIdx.x & 31;
    const int w    = threadIdx.x >> 5;              // 0..3
    const int m0   = blockIdx.x * 128 + (w & 1) * 64;
    const int n0   = blockIdx.y * 128 + (w >> 1) * 64;

    v8f acc[4][4];
#pragma unroll
    for (int i = 0; i < 4; ++i)
#pragma unroll
        for (int j = 0; j < 4; ++j) acc[i][j] = zero8();

    for (int kk = 0; kk < K; kk += 4) {
        v2f af[4], bf[4];
#pragma unroll
        for (int i = 0; i < 4; ++i) af[i] = frag_kcontig(A, K, m0 + 16 * i, kk, lane);
#pragma unroll
        for (int j = 0; j < 4; ++j) bf[j] = frag_kcontig(W, K, n0 + 16 * j, kk, lane);
#pragma unroll
        for (int i = 0; i < 4; ++i)
#pragma unroll
            for (int j = 0; j < 4; ++j) acc[i][j] = wmma4(af[i], bf[j], acc[i][j]);
    }

    const int rbase = ((lane >> 4) << 3);           // 0 or 8
    const int cl    = lane & 15;
#pragma unroll
    for (int j = 0; j < 4; ++j) {
        const int   col = n0 + 16 * j + cl;
        const float bv  = bias[col];
#pragma unroll
        for (int i = 0; i < 4; ++i) {
#pragma unroll
            for (int v = 0; v < 8; ++v) {
                const int row = m0 + 16 * i + rbase + v;
                C[(size_t)row * N + col] = acc[i][j][v] + bv;
            }
        }
    }
}

// ---------------------------------------------------------------------------
// Kernel 2: FUSED scores + head-softmax + att@V.
// One workgroup = 512 threads = 16 waves = one wave per head.
// Workgroup owns (b, 16-query tile); iterates over 32-key tiles of all S keys.
// Per iteration:
//   phase 1: wave h computes its 16x32 score tile (WMMA over dr=64), scales,
//            stores to LDS slab sS[h].
//   phase 2: 512 threads do softmax ACROSS HEADS for the 512 (q,k) pairs in
//            LDS, apply mask, write the final att values to d_out (only write).
//   phase 3: wave h consumes sS[h] as the WMMA A-operand against V_h,
//            accumulating z in registers (no rescaling needed: head-softmax
//            is local per (q,k)).
// ---------------------------------------------------------------------------
__global__ __launch_bounds__(512)
void fused_attn_k(const float* __restrict__ Q, const float* __restrict__ Kp,
                  const float* __restrict__ V, const int* __restrict__ mask,
                  float* __restrict__ att, float* __restrict__ Z) {
    const int lane = threadIdx.x & 31;
    const int h    = threadIdx.x >> 5;              // wave id == head id
    const int b    = blockIdx.y;
    const int q0   = blockIdx.x * QT;

    __shared__ float sS[NH][QT * KT];               // 16 * 512 * 4B = 32 KB

    const float* Qh = Q + (size_t)b * SS * EMB + h * DR;   // ld = EMB
    const float* Kh = Kp + (size_t)b * SS * EMB + h * DR;  // ld = EMB
    const float* Vh = V + (size_t)b * SS * EMB + h * DR;   // ld = EMB

    const int rbase = ((lane >> 4) << 3);           // 0 or 8
    const int cl    = lane & 15;
    float* sh = &sS[h][0];

    // Preload the Q fragments for this head's 16-query tile (K = dr = 64).
    v2f qf[16];
#pragma unroll
    for (int ks = 0; ks < 16; ++ks) qf[ks] = frag_kcontig(Qh, EMB, q0, 4 * ks, lane);

    v8f zacc[4];
#pragma unroll
    for (int j = 0; j < 4; ++j) zacc[j] = zero8();

    for (int kt = 0; kt < SS / KT; ++kt) {
        const int k0 = kt * KT;

        // ---- phase 1: score tile S_h[16q][32k] = 0.25 * Q_h K_h^T ----
        v8f sc[2];
        sc[0] = zero8(); sc[1] = zero8();
#pragma unroll
        for (int ks = 0; ks < 16; ++ks) {
            v2f b0 = frag_kcontig(Kh, EMB, k0,      4 * ks, lane);
            v2f b1 = frag_kcontig(Kh, EMB, k0 + 16, 4 * ks, lane);
            sc[0] = wmma4(qf[ks], b0, sc[0]);
            sc[1] = wmma4(qf[ks], b1, sc[1]);
        }
#pragma unroll
        for (int kj = 0; kj < 2; ++kj)
#pragma unroll
            for (int v = 0; v < 8; ++v)
                sh[(rbase + v) * KT + kj * 16 + cl] = sc[kj][v] * 0.25f;
        __syncthreads();

        // ---- phase 2: softmax over the 16 heads, per (q,k) pair ----
        {
            const int p  = threadIdx.x;             // 512 pairs exactly
            const int qq = p >> 5;                  // 0..15
            const int kk = p & 31;                  // 0..31
            float x[NH];
            if (mask[(size_t)b * PLANE + (size_t)(q0 + qq) * SS + (k0 + kk)] == 0) {
                // all heads masked to -1e11 -> uniform softmax over heads
#pragma unroll
                for (int hh = 0; hh < NH; ++hh) x[hh] = 1.0f / 16.0f;
            } else {
                float mx = -3.4e38f;
#pragma unroll
                for (int hh = 0; hh < NH; ++hh) {
                    x[hh] = sS[hh][p];
                    mx = fmaxf(mx, x[hh]);
                }
                float s = 0.f;
#pragma unroll
                for (int hh = 0; hh < NH; ++hh) {
                    x[hh] = __expf(x[hh] - mx);
                    s += x[hh];
                }
                const float inv = 1.f / s;
#pragma unroll
                for (int hh = 0; hh < NH; ++hh) x[hh] *= inv;
            }
#pragma unroll
            for (int hh = 0; hh < NH; ++hh) {
                sS[hh][p] = x[hh];
                att[(size_t)(b * NH + hh) * PLANE + (size_t)(q0 + qq) * SS + (k0 + kk)] = x[hh];
            }
        }
        __syncthreads();

        // ---- phase 3: z += att_tile(16x32, LDS) @ V_h(32x64, global) ----
#pragma unroll
        for (int ks2 = 0; ks2 < KT / 4; ++ks2) {
            const int kk4 = 4 * ks2;
            const int kA  = kk4 + ((lane >> 4) << 1);
            v2f a;
            a.x = sh[cl * KT + kA];
            a.y = sh[cl * KT + kA + 1];
#pragma unroll
            for (int dj = 0; dj < 4; ++dj) {
                v2f bf = frag_b_nn(Vh, EMB, 16 * dj, k0 + kk4, lane);
                zacc[dj] = wmma4(a, bf, zacc[dj]);
            }
        }
        // no extra barrier: only wave h touches sS[h] in phases 1/3, and the
        // phase-2 cross-wave accesses are fenced by the two barriers above.
    }

    // ---- write z (merged heads): Z[b, q0+row, h*64 + d] ----
#pragma unroll
    for (int dj = 0; dj < 4; ++dj)
#pragma unroll
        for (int v = 0; v < 8; ++v)
            Z[(size_t)b * SS * EMB + (size_t)(q0 + rbase + v) * EMB + h * DR + 16 * dj + cl]
                = zacc[dj][v];
}

// ---------------------------------------------------------------------------
extern "C" void kernel_launch(void* const* d_in, const int* in_sizes, int n_in,
                              void* d_out, int out_size, void* d_ws, size_t ws_size,
                              hipStream_t stream) {
    (void)in_sizes; (void)n_in; (void)out_size; (void)ws_size;

    const float* q    = (const float*)d_in[0];
    const float* k    = (const float*)d_in[1];
    const float* v    = (const float*)d_in[2];
    const int*   mask = (const int*)  d_in[3];
    const float* Wq   = (const float*)d_in[4];
    const float* bq   = (const float*)d_in[5];
    const float* Wk   = (const float*)d_in[6];
    const float* bk   = (const float*)d_in[7];
    const float* Wv   = (const float*)d_in[8];
    const float* bv   = (const float*)d_in[9];
    const float* Wo   = (const float*)d_in[10];
    const float* bo   = (const float*)d_in[11];

    float* out = (float*)d_out;                       // [B,S,E]   = 8M floats
    float* att = out + (size_t)BB * SS * EMB;         // [B,H,S,S] = 128M floats

    const size_t NQKV = (size_t)BB * SS * EMB;        // 8388608 elems (32 MB)
    float* Qb = (float*)d_ws;
    float* Kb = Qb + NQKV;
    float* Vb = Kb + NQKV;
    float* Zb = Vb + NQKV;                            // z needs its own slab now

    dim3 gP(MM / 128, EMB / 128);                     // (64, 8)
    gemm_nt_bias_k<<<gP, 128, 0, stream>>>(q, Wq, bq, Qb, EMB, EMB);
    gemm_nt_bias_k<<<gP, 128, 0, stream>>>(k, Wk, bk, Kb, EMB, EMB);
    gemm_nt_bias_k<<<gP, 128, 0, stream>>>(v, Wv, bv, Vb, EMB, EMB);

    fused_attn_k<<<dim3(SS / QT, BB), 512, 0, stream>>>(Qb, Kb, Vb, mask, att, Zb);

    gemm_nt_bias_k<<<gP, 128, 0, stream>>>(Zb, Wo, bo, out, EMB, EMB);
}